// PointMultiGraspNet_V3_58858231824616
// MI455X (gfx1250) — compile-verified
//
#include <hip/hip_runtime.h>
#include <hip/hip_bf16.h>

// ---------------------------------------------------------------------------
// PointMultiGraspNet_V3 forward for MI455X (gfx1250, wave32).
//  K1: coalesced NON-TEMPORAL b128 segment-max over (65536 x 1024) f32
//      -> features (256x1024). Only bandwidth-relevant stage:
//      256 MB @ 23.3 TB/s ~ 11 us; NT hint keeps the one-shot stream from
//      evicting the L2-resident weights used by the WMMA stages.
//  K2: f32 WMMA GEMM  Y = X * W^T + bias   (V_WMMA_F32_16X16X4_F32)
//  K3: LayerNorm+ReLU over 512 + info Linear(3->32) concat -> x (256x544)
//  K2 again (x2): anchor/offset hidden layers (544->256)
//  K5: per-row LayerNorm+ReLU + tiny head projections (256->6, 256->18)
// ---------------------------------------------------------------------------

#define SEG_B   256
#define FEAT_C  1024
#define K_PAD   512
#define LN_EPS  1e-5f

typedef __attribute__((ext_vector_type(2))) float v2f;
typedef __attribute__((ext_vector_type(4))) float v4f;
typedef __attribute__((ext_vector_type(8))) float v8f;

__device__ __forceinline__ v4f vmax4(v4f a, v4f b) {
    v4f r;
    r.x = fmaxf(a.x, b.x);
    r.y = fmaxf(a.y, b.y);
    r.z = fmaxf(a.z, b.z);
    r.w = fmaxf(a.w, b.w);
    return r;
}

// --------------------------- K1: segment max -------------------------------
// grid = 256 segments, block = 256 threads; thread t owns channels 4t..4t+3.
// Per wave each iteration issues b128 loads covering 512 contiguous bytes.
// Row loop unrolled x2 with independent accumulators for load pipelining.
__global__ void segmax_kernel(const float* __restrict__ feat,
                              const long long* __restrict__ offsets,
                              float* __restrict__ features) {
    const int b  = blockIdx.x;
    const int c4 = threadIdx.x << 2;                  // channel base (0..1020)
    const long long start = (b == 0) ? 0 : offsets[b - 1];
    const long long end   = offsets[b];
    const long long len   = end - start;

    v4f v0 = { -INFINITY, -INFINITY, -INFINITY, -INFINITY };
    v4f v1 = v0;
    const float* base = feat + c4;

    long long i = start;
    for (; i + 2 <= end; i += 2) {
        v4f a = __builtin_nontemporal_load((const v4f*)(base + i * (long long)FEAT_C));
        v4f c = __builtin_nontemporal_load((const v4f*)(base + (i + 1) * (long long)FEAT_C));
        v0 = vmax4(v0, a);
        v1 = vmax4(v1, c);
    }
    if (i < end) {
        v4f a = __builtin_nontemporal_load((const v4f*)(base + i * (long long)FEAT_C));
        v0 = vmax4(v0, a);
    }
    v0 = vmax4(v0, v1);

    // reference zero-pads segments shorter than K_PAD slots before the max
    if (len < K_PAD) {
        v4f z = { 0.f, 0.f, 0.f, 0.f };
        v0 = vmax4(v0, z);
    }
    *(v4f*)(features + (size_t)b * FEAT_C + c4) = v0;
}

// ------------------- K2: f32 WMMA GEMM  Y = X*W^T + bias -------------------
// X: (M x Kd) row-major, W: (N x Kd) row-major (i.e. we compute X @ W^T).
// One wave per 16x16 output tile; K swept in steps of 4 via
// V_WMMA_F32_16X16X4_F32. A/B fragments are contiguous float2 loads.
__global__ void wmma_gemm_kernel(const float* __restrict__ X,
                                 const float* __restrict__ W,
                                 const float* __restrict__ bias,
                                 float* __restrict__ Y,
                                 int M, int N, int Kd) {
    const int lane   = threadIdx.x & 31;
    const int wave   = blockIdx.x * (blockDim.x >> 5) + (threadIdx.x >> 5);
    const int tilesN = N >> 4;
    const int tm     = (wave / tilesN) << 4;
    const int tn     = (wave % tilesN) << 4;

    const int mrow  = tm + (lane & 15);        // A: M index for this lane
    const int ncol  = tn + (lane & 15);        // B: N index for this lane
    const int khalf = (lane >> 4) << 1;        // lanes 0-15 -> K=0,1 ; 16-31 -> K=2,3

    const float* xp = X + (size_t)mrow * Kd + khalf;
    const float* wp = W + (size_t)ncol * Kd + khalf;

    v8f acc = {};
#pragma unroll 4
    for (int k0 = 0; k0 < Kd; k0 += 4) {
        v2f a = *(const v2f*)(xp + k0);        // X[mrow][k0+khalf .. +1]
        v2f b = *(const v2f*)(wp + k0);        // W[ncol][k0+khalf .. +1] == B[k][n]
        acc = __builtin_amdgcn_wmma_f32_16x16x4_f32(
            /*neg_a=*/false, a, /*neg_b=*/false, b,
            /*c_mod=*/(short)0, acc, /*reuse_a=*/false, /*reuse_b=*/false);
    }

    // C/D layout: VGPR r -> M = tm + r (lanes 0-15) or tm + 8 + r (lanes 16-31)
    const int orow = tm + ((lane >> 4) << 3);
    const float bv = bias[ncol];
#pragma unroll
    for (int r = 0; r < 8; ++r) {
        Y[(size_t)(orow + r) * N + ncol] = acc[r] + bv;
    }
}

// ---------- K3: LN(512)+ReLU on point features, info linear, concat --------
// One block (256 threads) per batch row. Each thread owns cols t and t+256.
__global__ void ln_relu_concat_kernel(const float* __restrict__ pf_pre,
                                      const float* __restrict__ g,
                                      const float* __restrict__ bb,
                                      const float* __restrict__ info,
                                      const float* __restrict__ Wi,
                                      const float* __restrict__ bi,
                                      float* __restrict__ x) {
    __shared__ float s_sum[256];
    __shared__ float s_sq[256];
    const int row = blockIdx.x;
    const int t   = threadIdx.x;

    const float v0 = pf_pre[row * 512 + t];
    const float v1 = pf_pre[row * 512 + t + 256];
    s_sum[t] = v0 + v1;
    s_sq[t]  = v0 * v0 + v1 * v1;
    __syncthreads();
    for (int s = 128; s > 0; s >>= 1) {
        if (t < s) { s_sum[t] += s_sum[t + s]; s_sq[t] += s_sq[t + s]; }
        __syncthreads();
    }
    const float mean = s_sum[0] * (1.0f / 512.0f);
    const float var  = s_sq[0] * (1.0f / 512.0f) - mean * mean;
    const float inv  = rsqrtf(var + LN_EPS);

    x[row * 544 + t]       = fmaxf((v0 - mean) * inv * g[t]       + bb[t],       0.0f);
    x[row * 544 + t + 256] = fmaxf((v1 - mean) * inv * g[t + 256] + bb[t + 256], 0.0f);

    if (t < 32) {  // info_layer: Linear(3 -> 32), no LN/ReLU in reference
        float acc = bi[t];
#pragma unroll
        for (int k = 0; k < 3; ++k) acc += info[row * 3 + k] * Wi[t * 3 + k];
        x[row * 544 + 512 + t] = acc;
    }
}

// -------- K5: per-row LN(256)+ReLU for both heads + final projections ------
__global__ void heads_kernel(const float* __restrict__ ha,
                             const float* __restrict__ ho,
                             const float* __restrict__ ga, const float* __restrict__ bal,
                             const float* __restrict__ go, const float* __restrict__ bol,
                             const float* __restrict__ Wa2, const float* __restrict__ ba2,
                             const float* __restrict__ Wo2, const float* __restrict__ bo2,
                             float* __restrict__ pred, float* __restrict__ off) {
    __shared__ float r0[256], r1[256], r2[256], r3[256];
    __shared__ float na[256], no[256];
    const int row = blockIdx.x;
    const int t   = threadIdx.x;

    const float a = ha[row * 256 + t];
    const float o = ho[row * 256 + t];
    r0[t] = a; r1[t] = a * a; r2[t] = o; r3[t] = o * o;
    __syncthreads();
    for (int s = 128; s > 0; s >>= 1) {
        if (t < s) {
            r0[t] += r0[t + s]; r1[t] += r1[t + s];
            r2[t] += r2[t + s]; r3[t] += r3[t + s];
        }
        __syncthreads();
    }
    const float ma = r0[0] * (1.0f / 256.0f);
    const float va = r1[0] * (1.0f / 256.0f) - ma * ma;
    const float mo = r2[0] * (1.0f / 256.0f);
    const float vo = r3[0] * (1.0f / 256.0f) - mo * mo;

    na[t] = fmaxf((a - ma) * rsqrtf(va + LN_EPS) * ga[t] + bal[t], 0.0f);
    no[t] = fmaxf((o - mo) * rsqrtf(vo + LN_EPS) * go[t] + bol[t], 0.0f);
    __syncthreads();

    if (t < 6) {                               // pred: (256 -> 6)
        float s = ba2[t];
        for (int k = 0; k < 256; ++k) s += na[k] * Wa2[t * 256 + k];
        pred[row * 6 + t] = s;
    }
    if (t >= 32 && t < 50) {                   // offset: (256 -> 18)
        const int j = t - 32;
        float s = bo2[j];
        for (int k = 0; k < 256; ++k) s += no[k] * Wo2[j * 256 + k];
        off[row * 18 + j] = s;
    }
}

// ---------------------------------------------------------------------------
extern "C" void kernel_launch(void* const* d_in, const int* in_sizes, int n_in,
                              void* d_out, int out_size, void* d_ws, size_t ws_size,
                              hipStream_t stream) {
    (void)in_sizes; (void)n_in; (void)out_size; (void)ws_size;

    const float*     feat    = (const float*)d_in[0];
    const float*     info    = (const float*)d_in[1];
    const long long* offsets = (const long long*)d_in[2];   // int64
    const float*     W1      = (const float*)d_in[3];
    const float*     b1      = (const float*)d_in[4];
    const float*     ln1_g   = (const float*)d_in[5];
    const float*     ln1_b   = (const float*)d_in[6];
    const float*     Wi      = (const float*)d_in[7];
    const float*     bi      = (const float*)d_in[8];
    const float*     Wa1     = (const float*)d_in[9];
    const float*     ba1     = (const float*)d_in[10];
    const float*     lna_g   = (const float*)d_in[11];
    const float*     lna_b   = (const float*)d_in[12];
    const float*     Wa2     = (const float*)d_in[13];
    const float*     ba2     = (const float*)d_in[14];
    const float*     Wo1     = (const float*)d_in[15];
    const float*     bo1     = (const float*)d_in[16];
    const float*     lno_g   = (const float*)d_in[17];
    const float*     lno_b   = (const float*)d_in[18];
    const float*     Wo2     = (const float*)d_in[19];
    const float*     bo2     = (const float*)d_in[20];

    float* out      = (float*)d_out;
    float* features = out;                              // 256*1024
    float* pred     = out + SEG_B * FEAT_C;             // 256*6
    float* off      = out + SEG_B * FEAT_C + SEG_B * 6; // 256*18

    float* ws     = (float*)d_ws;
    float* pf_pre = ws;                       // 256*512 = 131072 floats
    float* x      = ws + 131072;              // 256*544 = 139264 floats
    float* ha     = ws + 131072 + 139264;     // 256*256 =  65536 floats
    float* ho     = ha + 65536;               // 256*256 =  65536 floats

    // K1: segment max (bandwidth stage: streams all 256 MB of feat once, NT)
    segmax_kernel<<<SEG_B, 256, 0, stream>>>(feat, offsets, features);

    // K2: point_layer GEMM  (256x1024)@(1024x512): 512 tiles -> 64 blocks x 8 waves
    wmma_gemm_kernel<<<64, 256, 0, stream>>>(features, W1, b1, pf_pre, 256, 512, 1024);

    // K3: LN + ReLU + info linear + concat -> x (256x544)
    ln_relu_concat_kernel<<<SEG_B, 256, 0, stream>>>(pf_pre, ln1_g, ln1_b, info, Wi, bi, x);

    // K2 again: head hidden layers (256x544)@(544x256): 256 tiles -> 32 blocks each
    wmma_gemm_kernel<<<32, 256, 0, stream>>>(x, Wa1, ba1, ha, 256, 256, 544);
    wmma_gemm_kernel<<<32, 256, 0, stream>>>(x, Wo1, bo1, ho, 256, 256, 544);

    // K5: per-row LN+ReLU + tiny output projections
    heads_kernel<<<SEG_B, 256, 0, stream>>>(ha, ho, lna_g, lna_b, lno_g, lno_b,
                                            Wa2, ba2, Wo2, bo2, pred, off);
}